// ROIPooler_87179246174701
// MI455X (gfx1250) — compile-verified
//
#include <hip/hip_runtime.h>

#define TH_ 7
#define TW_ 7
#define B_  16
#define H_  256
#define W_  256
#define C_  128
#define NROI_ 8
#define WMAX_ 16   // window width is <= 15 given roi ranges; pad to 16 for fixed async counts

// Low 32 bits of a generic pointer to __shared__ data are the LDS byte offset
// (flat->LDS mapping truncates to addr[31:0]) — exactly what the async-load
// VDST operand wants.
__device__ __forceinline__ unsigned lds_byte_off(const void* p) {
    return (unsigned)(unsigned long long)p;
}

__device__ __forceinline__ void async_copy_col_b128(unsigned lds_off, const float* gaddr) {
    // copies 16B/lane (512B/wave) from global address gaddr to LDS byte offset lds_off; tracked by ASYNCcnt
    asm volatile("global_load_async_to_lds_b128 %0, %1, off"
                 :: "v"(lds_off), "v"(gaddr)
                 : "memory");
}

__device__ __forceinline__ void wait_dscnt0() {
    // Retire all outstanding DS reads before async stores may overwrite their buffer.
    asm volatile("s_wait_dscnt 0x0" ::: "memory");
}

__global__ __launch_bounds__(32) void ROIPooler_87179246174701_kernel(
    const float* __restrict__ fm,    // [B,H,W,C]
    const float* __restrict__ rois,  // [B,NROI,4]
    float* __restrict__ out)         // [B,TH,TW,C]
{
    __shared__ __align__(16) float buf[3][WMAX_ * C_];   // 3 x 8KB row buffers

    const int cell = blockIdx.x;             // 0 .. B*TH*TW-1
    const int b    = cell / (TH_ * TW_);
    const int rr   = cell % (TH_ * TW_);
    const int th   = rr / TW_;
    const int tw   = rr % TW_;
    const int lane = threadIdx.x;            // 0..31, 4 channels per lane

    // Only roi index 0 of the NROI per batch is used by the reference.
    const float* roi = rois + (size_t)b * NROI_ * 4;
    const float y0 = roi[0], x0 = roi[1], y1 = roi[2], x1 = roi[3];

    // Replicate jnp f32 arithmetic (mul, sub, div, add; no contraction; trunc casts).
    const float y_start = __fmul_rn((float)H_, y0);
    const float x_start = __fmul_rn((float)W_, x0);
    const float h_step  = __fdiv_rn(__fmul_rn((float)H_, __fsub_rn(y1, y0)), (float)TH_);
    const float w_step  = __fdiv_rn(__fmul_rn((float)W_, __fsub_rn(x1, x0)), (float)TW_);

    const float hs_f = __fadd_rn(y_start, __fmul_rn((float)th, h_step));
    int hs = (int)hs_f;
    int he = (int)__fadd_rn(hs_f, h_step);
    he = hs + max(1, he - hs);

    const float ws_f = __fadd_rn(x_start, __fmul_rn((float)tw, w_step));
    int ws = (int)ws_f;
    int we = (int)__fadd_rn(ws_f, w_step);
    we = ws + max(1, we - ws);

    const int hn = min(he, H_) - hs;                    // valid rows
    const int wn = min(min(we, W_) - ws, WMAX_);        // valid cols (<= 15)

    float4 m;
    m.x = -__builtin_inff(); m.y = -__builtin_inff();
    m.z = -__builtin_inff(); m.w = -__builtin_inff();

    if (hn > 0 && wn > 0 && hs >= 0 && hs < H_ && ws >= 0 && ws < W_) {
        // Issue one row: 16 fixed async column copies (addresses clamped in-bounds,
        // invalid columns masked at consume time). Fixed count => compile-time
        // s_wait_asynccnt thresholds.
        auto issue_row = [&](int h, int bufidx) {
            const int hc = min(h, H_ - 1);
            const float* gbase =
                fm + ((size_t)(b * H_ + hc) * W_) * C_ + lane * 4;
            const unsigned lbase = lds_byte_off(&buf[bufidx][lane * 4]);
            #pragma unroll
            for (int j = 0; j < WMAX_; ++j) {
                const int wc = min(ws + j, W_ - 1);
                async_copy_col_b128(lbase + (unsigned)(j * C_ * 4),
                                    gbase + (size_t)wc * C_);
            }
        };

        // Triple-buffered software pipeline on ASYNCcnt (<=48 outstanding <= 63 max).
        issue_row(hs, 0);
        if (hn > 1) issue_row(hs + 1, 1);

        for (int hh = 0; hh < hn; ++hh) {
            if (hh + 2 < hn) {
                // buf[(hh+2)%3] was consumed at iteration hh-1; make sure those
                // DS reads retired before async stores may land in it.
                wait_dscnt0();
                issue_row(hs + hh + 2, (hh + 2) % 3);
            }

            // Retire exactly the row we are about to consume:
            // async loads complete in order; 16 issues per row.
            if (hh + 2 < hn) {
                asm volatile("s_wait_asynccnt 0x20" ::: "memory");  // 2 newer rows may remain
            } else if (hh + 1 < hn) {
                asm volatile("s_wait_asynccnt 0x10" ::: "memory");  // 1 newer row may remain
            } else {
                asm volatile("s_wait_asynccnt 0x0" ::: "memory");   // last row
            }

            const float* base = &buf[hh % 3][lane * 4];
            int j = 0;
            for (; j + 2 <= wn; j += 2) {                       // 2x unrolled ds_load_b128
                const float4 v0 = *(const float4*)(base + (size_t)j * C_);
                const float4 v1 = *(const float4*)(base + (size_t)(j + 1) * C_);
                m.x = fmaxf(m.x, v0.x); m.y = fmaxf(m.y, v0.y);
                m.z = fmaxf(m.z, v0.z); m.w = fmaxf(m.w, v0.w);
                m.x = fmaxf(m.x, v1.x); m.y = fmaxf(m.y, v1.y);
                m.z = fmaxf(m.z, v1.z); m.w = fmaxf(m.w, v1.w);
            }
            if (j < wn) {
                const float4 v = *(const float4*)(base + (size_t)j * C_);
                m.x = fmaxf(m.x, v.x); m.y = fmaxf(m.y, v.y);
                m.z = fmaxf(m.z, v.z); m.w = fmaxf(m.w, v.w);
            }
        }
    }

    // Coalesced 512B store per wave.
    *(float4*)(out + (size_t)cell * C_ + lane * 4) = m;
}

extern "C" void kernel_launch(void* const* d_in, const int* in_sizes, int n_in,
                              void* d_out, int out_size, void* d_ws, size_t ws_size,
                              hipStream_t stream) {
    (void)in_sizes; (void)n_in; (void)d_ws; (void)ws_size; (void)out_size;
    const float* fm   = (const float*)d_in[0];
    const float* rois = (const float*)d_in[1];
    float* out        = (float*)d_out;

    dim3 grid(B_ * TH_ * TW_);   // 784 wave32 blocks, one per output cell
    dim3 block(32);
    hipLaunchKernelGGL(ROIPooler_87179246174701_kernel, grid, block, 0, stream,
                       fm, rois, out);
}